// GCN_19155554140397
// MI455X (gfx1250) — compile-verified
//
#include <hip/hip_runtime.h>

typedef __attribute__((ext_vector_type(2))) float v2f;
typedef __attribute__((ext_vector_type(8))) float v8f;

#define CIN  128
#define CHID 64
#define COUT 40
#define COUTP 48   // COUT padded to 3x16 for WMMA tiles

// ---------------- degree / norm ----------------
__global__ void k_init_deg(float* __restrict__ deg, int n) {
  int i = blockIdx.x * blockDim.x + threadIdx.x;
  if (i < n) deg[i] = 1.0f;  // self loop
}

__global__ void k_deg_accum(const int* __restrict__ dst, float* __restrict__ deg, int e) {
  int i = blockIdx.x * blockDim.x + threadIdx.x;
  if (i < e)
    __hip_atomic_fetch_add(&deg[dst[i]], 1.0f, __ATOMIC_RELAXED, __HIP_MEMORY_SCOPE_AGENT);
}

__global__ void k_rsqrt(float* __restrict__ deg, int n) {
  int i = blockIdx.x * blockDim.x + threadIdx.x;
  if (i < n) deg[i] = rsqrtf(deg[i]);
}

// ---------------- GEMM1: H1[n,64] = feat[n,128] @ W1[128,64] (WMMA f32 16x16x4) ----------------
// LDS holds W1 K-pair interleaved: sW[(k>>1)*(2*CHID) + 2*n + (k&1)] = W[k][n]
// => B fragment {W[kk][n], W[kk+1][n]} is one aligned ds_load_b64.
__global__ void __launch_bounds__(256) k_gemm1(const float* __restrict__ A,
                                               const float* __restrict__ W,
                                               float* __restrict__ H, int ntiles) {
  __shared__ float sW[CIN * CHID];  // 32 KB
  int tid = threadIdx.x;
  for (int i = tid; i < CIN * CHID; i += 256) {
    int k = i >> 6;           // i / CHID
    int c = i & (CHID - 1);   // i % CHID
    sW[(k >> 1) * (2 * CHID) + 2 * c + (k & 1)] = W[i];
  }
  __syncthreads();

  int wave = tid >> 5;
  int lane = tid & 31;
  int half = lane >> 4;   // K sub-group for A/B fragments
  int l16  = lane & 15;   // M index (A) / N index (B,C)
  int tile = blockIdx.x * 8 + wave;
  if (tile >= ntiles) return;            // wave-uniform: EXEC all-1s inside WMMA

  int m0 = tile * 16;
  v8f acc0 = {}, acc1 = {}, acc2 = {}, acc3 = {};
  const float* arow = A + (size_t)(m0 + l16) * CIN;

#pragma unroll 4
  for (int k = 0; k < CIN; k += 4) {
    int kk = k + 2 * half;
    // A 16x4 f32 fragment: v0 = A[m][kk], v1 = A[m][kk+1]
    v2f a = *(const v2f*)(arow + kk);
    const float* bp = sW + (kk >> 1) * (2 * CHID) + 2 * l16;
    v2f b0 = *(const v2f*)(bp +  0);
    v2f b1 = *(const v2f*)(bp + 32);
    v2f b2 = *(const v2f*)(bp + 64);
    v2f b3 = *(const v2f*)(bp + 96);
    acc0 = __builtin_amdgcn_wmma_f32_16x16x4_f32(false, a, false, b0, (short)0, acc0, false, false);
    acc1 = __builtin_amdgcn_wmma_f32_16x16x4_f32(false, a, false, b1, (short)0, acc1, false, false);
    acc2 = __builtin_amdgcn_wmma_f32_16x16x4_f32(false, a, false, b2, (short)0, acc2, false, false);
    acc3 = __builtin_amdgcn_wmma_f32_16x16x4_f32(false, a, false, b3, (short)0, acc3, false, false);
  }

  // C/D layout: VGPR r, lanes[half]: row = m0 + r + 8*half, col = nt*16 + l16
  float* hrow = H + (size_t)(m0 + 8 * half) * CHID + l16;
#pragma unroll
  for (int r = 0; r < 8; ++r) {
    hrow[(size_t)r * CHID +  0] = acc0[r];
    hrow[(size_t)r * CHID + 16] = acc1[r];
    hrow[(size_t)r * CHID + 32] = acc2[r];
    hrow[(size_t)r * CHID + 48] = acc3[r];
  }
}

// ---------------- GEMM2: H2[n,40] = X1[n,64] @ W2[64,40] (N padded to 48) ----------------
__global__ void __launch_bounds__(256) k_gemm2(const float* __restrict__ X,
                                               const float* __restrict__ W,
                                               float* __restrict__ H, int ntiles) {
  __shared__ float sW[CHID * COUTP];  // 12 KB, K-pair interleaved, zero-padded cols 40..47
  int tid = threadIdx.x;
  for (int i = tid; i < CHID * COUTP; i += 256) {
    int k = i / COUTP, c = i - k * COUTP;
    float v = (c < COUT) ? W[k * COUT + c] : 0.0f;
    sW[(k >> 1) * (2 * COUTP) + 2 * c + (k & 1)] = v;
  }
  __syncthreads();

  int wave = tid >> 5;
  int lane = tid & 31;
  int half = lane >> 4;
  int l16  = lane & 15;
  int tile = blockIdx.x * 8 + wave;
  if (tile >= ntiles) return;

  int m0 = tile * 16;
  v8f acc0 = {}, acc1 = {}, acc2 = {};
  const float* xrow = X + (size_t)(m0 + l16) * CHID;

#pragma unroll 4
  for (int k = 0; k < CHID; k += 4) {
    int kk = k + 2 * half;
    v2f a = *(const v2f*)(xrow + kk);
    const float* bp = sW + (kk >> 1) * (2 * COUTP) + 2 * l16;
    v2f b0 = *(const v2f*)(bp +  0);
    v2f b1 = *(const v2f*)(bp + 32);
    v2f b2 = *(const v2f*)(bp + 64);
    acc0 = __builtin_amdgcn_wmma_f32_16x16x4_f32(false, a, false, b0, (short)0, acc0, false, false);
    acc1 = __builtin_amdgcn_wmma_f32_16x16x4_f32(false, a, false, b1, (short)0, acc1, false, false);
    acc2 = __builtin_amdgcn_wmma_f32_16x16x4_f32(false, a, false, b2, (short)0, acc2, false, false);
  }

  float* hrow = H + (size_t)(m0 + 8 * half) * COUT;
#pragma unroll
  for (int r = 0; r < 8; ++r) {
    hrow[(size_t)r * COUT +  0 + l16] = acc0[r];
    hrow[(size_t)r * COUT + 16 + l16] = acc1[r];
    if (l16 < COUT - 32) hrow[(size_t)r * COUT + 32 + l16] = acc2[r];
  }
}

// ---------------- aggregation: self loops + edge scatter-add ----------------
__global__ void k_self64(const float* __restrict__ H, const float* __restrict__ dinv,
                         float* __restrict__ X, int n) {
  int idx = blockIdx.x * blockDim.x + threadIdx.x;
  if (idx >= n * 16) return;
  int i = idx >> 4, q = idx & 15;
  float w = dinv[i] * dinv[i];
  float4 h = *(const float4*)(H + (size_t)i * CHID + q * 4);
  *(float4*)(X + (size_t)i * CHID + q * 4) = make_float4(w * h.x, w * h.y, w * h.z, w * h.w);
}

__global__ void k_edge64(const int* __restrict__ src, const int* __restrict__ dst,
                         const float* __restrict__ dinv, const float* __restrict__ H,
                         float* __restrict__ X, int e) {
  int idx = blockIdx.x * blockDim.x + threadIdx.x;
  if (idx >= e * 16) return;
  int ed = idx >> 4, q = idx & 15;
  int s = src[ed], d = dst[ed];
  float w = dinv[s] * dinv[d];
  float4 h = *(const float4*)(H + (size_t)s * CHID + q * 4);
  float* o = X + (size_t)d * CHID + q * 4;
  __hip_atomic_fetch_add(o + 0, w * h.x, __ATOMIC_RELAXED, __HIP_MEMORY_SCOPE_AGENT);
  __hip_atomic_fetch_add(o + 1, w * h.y, __ATOMIC_RELAXED, __HIP_MEMORY_SCOPE_AGENT);
  __hip_atomic_fetch_add(o + 2, w * h.z, __ATOMIC_RELAXED, __HIP_MEMORY_SCOPE_AGENT);
  __hip_atomic_fetch_add(o + 3, w * h.w, __ATOMIC_RELAXED, __HIP_MEMORY_SCOPE_AGENT);
}

__global__ void k_bias_relu(float* __restrict__ X, const float* __restrict__ b, int n) {
  int idx = blockIdx.x * blockDim.x + threadIdx.x;
  if (idx >= n * 16) return;
  int i = idx >> 4, q = idx & 15;
  float4 v  = *(float4*)(X + (size_t)i * CHID + q * 4);
  float4 bb = *(const float4*)(b + q * 4);
  v.x = fmaxf(v.x + bb.x, 0.0f);
  v.y = fmaxf(v.y + bb.y, 0.0f);
  v.z = fmaxf(v.z + bb.z, 0.0f);
  v.w = fmaxf(v.w + bb.w, 0.0f);
  *(float4*)(X + (size_t)i * CHID + q * 4) = v;
}

__global__ void k_self40(const float* __restrict__ H, const float* __restrict__ dinv,
                         float* __restrict__ O, int n) {
  int idx = blockIdx.x * blockDim.x + threadIdx.x;
  if (idx >= n * 10) return;
  int i = idx / 10, q = idx - i * 10;
  float w = dinv[i] * dinv[i];
  float4 h = *(const float4*)(H + (size_t)i * COUT + q * 4);
  *(float4*)(O + (size_t)i * COUT + q * 4) = make_float4(w * h.x, w * h.y, w * h.z, w * h.w);
}

__global__ void k_edge40(const int* __restrict__ src, const int* __restrict__ dst,
                         const float* __restrict__ dinv, const float* __restrict__ H,
                         float* __restrict__ O, int e) {
  int idx = blockIdx.x * blockDim.x + threadIdx.x;
  if (idx >= e * 10) return;
  int ed = idx / 10, q = idx - ed * 10;
  int s = src[ed], d = dst[ed];
  float w = dinv[s] * dinv[d];
  float4 h = *(const float4*)(H + (size_t)s * COUT + q * 4);
  float* o = O + (size_t)d * COUT + q * 4;
  __hip_atomic_fetch_add(o + 0, w * h.x, __ATOMIC_RELAXED, __HIP_MEMORY_SCOPE_AGENT);
  __hip_atomic_fetch_add(o + 1, w * h.y, __ATOMIC_RELAXED, __HIP_MEMORY_SCOPE_AGENT);
  __hip_atomic_fetch_add(o + 2, w * h.z, __ATOMIC_RELAXED, __HIP_MEMORY_SCOPE_AGENT);
  __hip_atomic_fetch_add(o + 3, w * h.w, __ATOMIC_RELAXED, __HIP_MEMORY_SCOPE_AGENT);
}

// ---------------- bias + log_softmax, one wave32 per row of 40 ----------------
__global__ void __launch_bounds__(256) k_logsoftmax(float* __restrict__ O,
                                                    const float* __restrict__ b2, int n) {
  int wave = threadIdx.x >> 5;
  int lane = threadIdx.x & 31;
  int row  = blockIdx.x * 8 + wave;
  if (row >= n) return;
  float* p = O + (size_t)row * COUT;
  float x0 = p[lane] + b2[lane];                                   // cols 0..31
  float x1 = (lane < 8) ? (p[32 + lane] + b2[32 + lane]) : -__builtin_inff();  // cols 32..39
  float mx = fmaxf(x0, x1);
#pragma unroll
  for (int off = 16; off > 0; off >>= 1) mx = fmaxf(mx, __shfl_xor(mx, off, 32));
  float s = __expf(x0 - mx) + ((lane < 8) ? __expf(x1 - mx) : 0.0f);
#pragma unroll
  for (int off = 16; off > 0; off >>= 1) s += __shfl_xor(s, off, 32);
  float l = mx + __logf(s);
  p[lane] = x0 - l;
  if (lane < 8) p[32 + lane] = x1 - l;
}

// ---------------- launcher ----------------
extern "C" void kernel_launch(void* const* d_in, const int* in_sizes, int n_in,
                              void* d_out, int out_size, void* d_ws, size_t ws_size,
                              hipStream_t stream) {
  const float* feat = (const float*)d_in[0];
  const int*   src  = (const int*)d_in[1];
  const int*   dst  = (const int*)d_in[2];
  const float* W1   = (const float*)d_in[3];
  const float* b1   = (const float*)d_in[4];
  const float* W2   = (const float*)d_in[5];
  const float* b2   = (const float*)d_in[6];
  float* out = (float*)d_out;

  const int n = in_sizes[0] / CIN;   // 50000 (multiple of 16)
  const int e = in_sizes[1];         // 800000

  float* dinv = (float*)d_ws;                     // n floats
  float* buf  = dinv + n;                         // n*64 floats (H1, then H2[n,40])
  float* x1   = buf + (size_t)n * CHID;           // n*64 floats

  const int B = 256;
  int ntiles = n / 16;

  k_init_deg <<<(n + B - 1) / B, B, 0, stream>>>(dinv, n);
  k_deg_accum<<<(e + B - 1) / B, B, 0, stream>>>(dst, dinv, e);
  k_rsqrt    <<<(n + B - 1) / B, B, 0, stream>>>(dinv, n);

  k_gemm1    <<<(ntiles + 7) / 8, B, 0, stream>>>(feat, W1, buf, ntiles);
  k_self64   <<<(n * 16 + B - 1) / B, B, 0, stream>>>(buf, dinv, x1, n);
  k_edge64   <<<(e * 16 + B - 1) / B, B, 0, stream>>>(src, dst, dinv, buf, x1, e);
  k_bias_relu<<<(n * 16 + B - 1) / B, B, 0, stream>>>(x1, b1, n);

  k_gemm2    <<<(ntiles + 7) / 8, B, 0, stream>>>(x1, W2, buf, ntiles);
  k_self40   <<<(n * 10 + B - 1) / B, B, 0, stream>>>(buf, dinv, out, n);
  k_edge40   <<<(e * 10 + B - 1) / B, B, 0, stream>>>(src, dst, dinv, buf, out, e);

  k_logsoftmax<<<(n + 7) / 8, B, 0, stream>>>(out, b2, n);
}